// GCN_19980187861533
// MI455X (gfx1250) — compile-verified
//
#include <hip/hip_runtime.h>
#include <hip/hip_bf16.h>

#define HIDDEN    256
#define NPG       200
#define NGRAPH    8
#define NEG_SLOPE 0.2f
#define BN_EPS    1e-5f

typedef __attribute__((ext_vector_type(2))) float v2f;
typedef __attribute__((ext_vector_type(8))) float v8f;

// ---------------------------------------------------------------- utilities
__global__ __launch_bounds__(256) void zero_kernel(float* __restrict__ p, int n) {
  int i = blockIdx.x * blockDim.x + threadIdx.x;
  if (i < n) p[i] = 0.0f;
}

__global__ __launch_bounds__(256) void init_nodes_kernel(float* __restrict__ deg,
                                                         float* __restrict__ cnt, int n) {
  int i = blockIdx.x * blockDim.x + threadIdx.x;
  if (i < n) { deg[i] = 1.0f; cnt[i] = 1.0f; }   // self-loop weight / self message
}

__global__ __launch_bounds__(256) void deg_accum_kernel(const int* __restrict__ dst,
                                                        const float* __restrict__ ea,
                                                        float* __restrict__ deg, int E) {
  int e = blockIdx.x * blockDim.x + threadIdx.x;
  if (e < E) atomicAdd(&deg[dst[e]], fabsf(ea[e]));
}

// deg -> dinv (in place); snorm gets self-loop term dinv^2; A gets diagonal self-loops
__global__ __launch_bounds__(256) void finalize_nodes_kernel(float* __restrict__ deg,
                                                             float* __restrict__ snorm,
                                                             float* __restrict__ Amat, int N) {
  int i = blockIdx.x * blockDim.x + threadIdx.x;
  if (i < N) {
    float d = deg[i];
    float di = d > 0.0f ? rsqrtf(d) : 0.0f;
    deg[i] = di;
    snorm[i] = di * di;
    int g = i / NPG, l = i - g * NPG;
    Amat[(size_t)g * NPG * NPG + (size_t)l * NPG + l] += 1.0f;  // Amat pre-zeroed
  }
}

__global__ __launch_bounds__(256) void edge_accum_kernel(const int* __restrict__ src,
                                                         const int* __restrict__ dst,
                                                         const float* __restrict__ ea,
                                                         const float* __restrict__ dinv,
                                                         float* __restrict__ Amat,
                                                         float* __restrict__ snorm,
                                                         float* __restrict__ cnt, int E) {
  int e = blockIdx.x * blockDim.x + threadIdx.x;
  if (e < E) {
    int s = src[e], d = dst[e];
    int g = d / NPG;
    atomicAdd(&Amat[(size_t)g * NPG * NPG + (size_t)(d - g * NPG) * NPG + (s - g * NPG)], 1.0f);
    atomicAdd(&snorm[d], dinv[s] * fabsf(ea[e]) * dinv[d]);
    atomicAdd(&cnt[d], 1.0f);
  }
}

// ------------------------------------------------- fp32 WMMA GEMM (batched)
// C[M,N] = act(A[M,K] @ B[K,N] + bias), row-major, lda=K, ldb=N.
// One wave -> 64(M) x 16(N) of C: 4 accumulators sharing one B fragment per
// K-step => 6 VMEM (4x b64 A + 2x b32 B) per 4 WMMAs.
// Block = 4 waves covering a 64x64 C tile.
__global__ __launch_bounds__(128)
void wmma_gemm_kernel(const float* __restrict__ A, const float* __restrict__ B,
                      float* __restrict__ C, const float* __restrict__ bias,
                      int M, int N, int K,
                      long long sA, long long sB, long long sC, int leaky) {
  const int lane  = threadIdx.x & 31;
  const int wave  = threadIdx.y;
  const int tileM = blockIdx.x << 6;                    // 64 rows per block
  const int tileN = (blockIdx.y * 4 + wave) << 4;       // 16 cols per wave
  if (tileN >= N) return;                               // wave-uniform
  A += (long long)blockIdx.z * sA;
  B += (long long)blockIdx.z * sB;
  C += (long long)blockIdx.z * sC;

  const int ncol = tileN + (lane & 15);
  const int kb   = (lane >> 4) << 1;  // lanes 0-15: K=0,1 ; lanes 16-31: K=2,3

  // A fragment rows for the 4 M-subtiles (clamped; garbage rows never stored).
  // Individually named pointers (not an array) so clang keeps them in the
  // global address space -> global_load_b64, not flat_load_b64.
  int m0 = tileM + (lane & 15);        if (m0 >= M) m0 = M - 1;
  int m1 = tileM + 16 + (lane & 15);   if (m1 >= M) m1 = M - 1;
  int m2 = tileM + 32 + (lane & 15);   if (m2 >= M) m2 = M - 1;
  int m3 = tileM + 48 + (lane & 15);   if (m3 >= M) m3 = M - 1;
  const float* Ap0 = A + (long long)m0 * K + kb;
  const float* Ap1 = A + (long long)m1 * K + kb;
  const float* Ap2 = A + (long long)m2 * K + kb;
  const float* Ap3 = A + (long long)m3 * K + kb;
  const float* Bp  = B + (long long)kb * N + ncol;

  v8f acc0 = {}, acc1 = {}, acc2 = {}, acc3 = {};
#pragma unroll 2
  for (int k0 = 0; k0 < K; k0 += 4) {                   // K % 4 == 0 for all calls
    v2f a0, a1, a2, a3, bv;
    a0.x = Ap0[0]; a0.y = Ap0[1];
    a1.x = Ap1[0]; a1.y = Ap1[1];
    a2.x = Ap2[0]; a2.y = Ap2[1];
    a3.x = Ap3[0]; a3.y = Ap3[1];
    bv.x = Bp[0];
    bv.y = Bp[N];
    Ap0 += 4; Ap1 += 4; Ap2 += 4; Ap3 += 4;
    Bp += (long long)4 * N;
    acc0 = __builtin_amdgcn_wmma_f32_16x16x4_f32(false, a0, false, bv, (short)0, acc0, false, false);
    acc1 = __builtin_amdgcn_wmma_f32_16x16x4_f32(false, a1, false, bv, (short)0, acc1, false, false);
    acc2 = __builtin_amdgcn_wmma_f32_16x16x4_f32(false, a2, false, bv, (short)0, acc2, false, false);
    acc3 = __builtin_amdgcn_wmma_f32_16x16x4_f32(false, a3, false, bv, (short)0, acc3, false, false);
  }

  const float bb   = bias ? bias[ncol] : 0.0f;
  const int   half = (lane >> 4) << 3;  // C VGPR v: M=v (lanes 0-15) / M=v+8
  v8f accs[4] = {acc0, acc1, acc2, acc3};
#pragma unroll
  for (int r = 0; r < 4; ++r) {
#pragma unroll
    for (int v = 0; v < 8; ++v) {
      int row = tileM + (r << 4) + half + v;
      if (row < M) {
        float val = accs[r][v] + bb;
        if (leaky) val = val >= 0.0f ? val : NEG_SLOPE * val;
        C[(long long)row * N + ncol] = val;
      }
    }
  }
}

// agg += cnt*(hi + be) + snorm*wlast + bg   (agg already holds A@hj)
__global__ __launch_bounds__(256)
void combine_kernel(float* __restrict__ agg, const float* __restrict__ hi,
                    const float* __restrict__ cnt, const float* __restrict__ snorm,
                    const float* __restrict__ be, const float* __restrict__ wlast,
                    const float* __restrict__ bg, int N) {
  int i = blockIdx.x * blockDim.x + threadIdx.x;
  if (i < N * HIDDEN) {
    int n = i >> 8;       // / HIDDEN
    int c = i & 255;      // % HIDDEN
    agg[i] = agg[i] + cnt[n] * (hi[i] + be[c]) + snorm[n] * wlast[c] + bg[c];
  }
}

// fused per-column BatchNorm (biased variance, training-forward semantics)
__global__ __launch_bounds__(256)
void bn_kernel(float* __restrict__ z, const float* __restrict__ gamma,
               const float* __restrict__ beta, int N) {
  __shared__ float s1[256], s2[256];
  const int c = blockIdx.x, t = threadIdx.x;
  float sum = 0.f, sq = 0.f;
  for (int r = t; r < N; r += 256) {
    float v = z[(size_t)r * HIDDEN + c];
    sum += v; sq += v * v;
  }
  s1[t] = sum; s2[t] = sq;
  __syncthreads();
  for (int s = 128; s > 0; s >>= 1) {
    if (t < s) { s1[t] += s1[t + s]; s2[t] += s2[t + s]; }
    __syncthreads();
  }
  float m     = s1[0] / (float)N;
  float var   = s2[0] / (float)N - m * m;
  float scale = gamma[c] * rsqrtf(var + BN_EPS);
  float shift = beta[c] - m * scale;
  for (int r = t; r < N; r += 256) {
    size_t idx = (size_t)r * HIDDEN + c;
    z[idx] = z[idx] * scale + shift;
  }
}

__global__ __launch_bounds__(256)
void pool_kernel(const float* __restrict__ z, const int* __restrict__ batch,
                 float* __restrict__ pooled, int N) {
  int i = blockIdx.x * blockDim.x + threadIdx.x;  // over NGRAPH*HIDDEN
  if (i < NGRAPH * HIDDEN) {
    int g = i / HIDDEN, c = i - g * HIDDEN;
    float s = 0.f, n = 0.f;
    for (int r = 0; r < N; ++r)
      if (batch[r] == g) { s += z[(size_t)r * HIDDEN + c]; n += 1.f; }
    pooled[i] = n > 0.f ? s / n : 0.f;
  }
}

__global__ __launch_bounds__(256)
void dense_kernel(const float* __restrict__ X, const float* __restrict__ W,
                  const float* __restrict__ b, float* __restrict__ Y,
                  int M, int N, int K, int leaky) {
  int i = blockIdx.x * blockDim.x + threadIdx.x;
  if (i < M * N) {
    int m = i / N, n = i - m * N;
    float s = b[n];
    for (int k = 0; k < K; ++k) s += X[(size_t)m * K + k] * W[(size_t)k * N + n];
    if (leaky) s = s >= 0.f ? s : NEG_SLOPE * s;
    Y[i] = s;
  }
}

// ------------------------------------------------------------------- driver
extern "C" void kernel_launch(void* const* d_in, const int* in_sizes, int n_in,
                              void* d_out, int out_size, void* d_ws, size_t ws_size,
                              hipStream_t stream) {
  const float* x     = (const float*)d_in[0];
  const int*   eidx  = (const int*)d_in[1];
  const float* ea    = (const float*)d_in[2];
  const int*   batch = (const int*)d_in[3];
  const int E      = in_sizes[2];
  const int N      = in_sizes[3];
  const int IN_DIM = in_sizes[0] / N;
  const int* src = eidx;
  const int* dst = eidx + E;

  // params flattened in setup_inputs() dict insertion order:
  // per layer: Wg, bg, We, be, W2, b2, gamma, beta
  const float *Wg[2], *bgp[2], *Wep[2], *bep[2], *W2p[2], *b2p[2], *gam[2], *bet[2];
  for (int L = 0; L < 2; ++L) {
    int base = 4 + L * 8;
    Wg[L]  = (const float*)d_in[base + 0];
    bgp[L] = (const float*)d_in[base + 1];
    Wep[L] = (const float*)d_in[base + 2];
    bep[L] = (const float*)d_in[base + 3];
    W2p[L] = (const float*)d_in[base + 4];
    b2p[L] = (const float*)d_in[base + 5];
    gam[L] = (const float*)d_in[base + 6];
    bet[L] = (const float*)d_in[base + 7];
  }
  const float* f1W = (const float*)d_in[20]; const float* f1b = (const float*)d_in[21];
  const float* f2W = (const float*)d_in[22]; const float* f2b = (const float*)d_in[23];
  const float* f3W = (const float*)d_in[24]; const float* f3b = (const float*)d_in[25];

  // workspace carve (~6.3 MB)
  float* ws = (float*)d_ws;
  size_t off = 0;
  auto carve = [&](size_t nElem) {
    float* p = ws + off;
    off += (nElem + 255) & ~(size_t)255;
    return p;
  };
  float* B0     = carve((size_t)N * HIDDEN);
  float* B1     = carve((size_t)N * HIDDEN);
  float* B2     = carve((size_t)N * HIDDEN);
  float* Amat   = carve((size_t)NGRAPH * NPG * NPG);
  float* dinv   = carve(N);                 // holds deg, then dinv
  float* cnt    = carve(N);
  float* snorm  = carve(N);
  float* pooled = carve(NGRAPH * HIDDEN);
  float* m1     = carve(NGRAPH * HIDDEN);
  float* m2     = carve(NGRAPH * 32);

  // ---- graph preprocessing (shared by both layers)
  const int asz = NGRAPH * NPG * NPG;
  zero_kernel<<<(asz + 255) / 256, 256, 0, stream>>>(Amat, asz);
  init_nodes_kernel<<<(N + 255) / 256, 256, 0, stream>>>(dinv, cnt, N);
  deg_accum_kernel<<<(E + 255) / 256, 256, 0, stream>>>(dst, ea, dinv, E);
  finalize_nodes_kernel<<<(N + 255) / 256, 256, 0, stream>>>(dinv, snorm, Amat, N);
  edge_accum_kernel<<<(E + 255) / 256, 256, 0, stream>>>(src, dst, ea, dinv, Amat, snorm, cnt, E);

  dim3 blk(32, 4, 1);
  auto gemm = [&](const float* A, const float* B, float* C, const float* bias,
                  int M, int Nc, int K, long long sA, long long sB, long long sC,
                  int nb, int leaky) {
    dim3 grid((M + 63) / 64, (Nc + 63) / 64, nb);
    wmma_gemm_kernel<<<grid, blk, 0, stream>>>(A, B, C, bias, M, Nc, K, sA, sB, sC, leaky);
  };

  // ---- GCN layers: h=B0, hi=B1, hj=B2, agg=B0(reuse), z=B2(reuse)
  const float* cur = x;
  int curK = IN_DIM;
  for (int L = 0; L < 2; ++L) {
    gemm(cur, Wg[L], B0, nullptr, N, HIDDEN, curK, 0, 0, 0, 1, 0);                     // h
    gemm(B0, Wep[L],                   B1, nullptr, N, HIDDEN, HIDDEN, 0, 0, 0, 1, 0); // hi = h@Wi
    gemm(B0, Wep[L] + HIDDEN * HIDDEN, B2, nullptr, N, HIDDEN, HIDDEN, 0, 0, 0, 1, 0); // hj = h@Wj
    // agg = A @ hj  (8 independent per-graph GEMMs), written over h
    gemm(Amat, B2, B0, nullptr, NPG, HIDDEN, NPG,
         (long long)NPG * NPG, (long long)NPG * HIDDEN, (long long)NPG * HIDDEN,
         NGRAPH, 0);
    combine_kernel<<<(N * HIDDEN + 255) / 256, 256, 0, stream>>>(
        B0, B1, cnt, snorm, bep[L], Wep[L] + 2 * HIDDEN * HIDDEN, bgp[L], N);
    gemm(B0, W2p[L], B2, b2p[L], N, HIDDEN, HIDDEN, 0, 0, 0, 1, 1);                    // z = leaky(agg@W2+b2)
    bn_kernel<<<HIDDEN, 256, 0, stream>>>(B2, gam[L], bet[L], N);
    cur = B2;
    curK = HIDDEN;
  }

  // ---- pool + MLP head
  pool_kernel<<<(NGRAPH * HIDDEN + 255) / 256, 256, 0, stream>>>(B2, batch, pooled, N);
  dense_kernel<<<(NGRAPH * 256 + 255) / 256, 256, 0, stream>>>(pooled, f1W, f1b, m1,
                                                               NGRAPH, 256, HIDDEN, 1);
  dense_kernel<<<1, 256, 0, stream>>>(m1, f2W, f2b, m2, NGRAPH, 32, 256, 1);
  dense_kernel<<<1, 32, 0, stream>>>(m2, f3W, f3b, (float*)d_out, NGRAPH, 2, 32, 0);
}